// BatchedHomoModel_22179211116720
// MI455X (gfx1250) — compile-verified
//
#include <hip/hip_runtime.h>

typedef __attribute__((ext_vector_type(2))) float v2f;
typedef __attribute__((ext_vector_type(8))) float v8f;

#define DIM 128

__device__ __forceinline__ float lrelu(float x) { return x > 0.0f ? x : 0.01f * x; }

__device__ __forceinline__ void atomic_fadd(float* p, float v) {
  __hip_atomic_fetch_add(p, v, __ATOMIC_RELAXED, __HIP_MEMORY_SCOPE_AGENT);
}

// ---------------- zero scratch (float4 grid-stride) ----------------
__global__ void k_zero(float4* __restrict__ p, long n4) {
  long i = (long)blockIdx.x * blockDim.x + threadIdx.x;
  long stride = (long)gridDim.x * blockDim.x;
  float4 z = {0.f, 0.f, 0.f, 0.f};
  for (; i < n4; i += stride) p[i] = z;
}

// ---------------- degree counting via float atomics ----------------
__global__ void k_degcount(const int* __restrict__ e1s, const int* __restrict__ e1d,
                           const int* __restrict__ e2s, const int* __restrict__ e2d,
                           int E1, int E2,
                           float* __restrict__ rout1, float* __restrict__ rin1,
                           float* __restrict__ rout2, float* __restrict__ rin2) {
  int i = blockIdx.x * blockDim.x + threadIdx.x;
  int stride = gridDim.x * blockDim.x;
  for (; i < E1; i += stride) {
    atomic_fadd(rout1 + e1s[i], 1.0f);
    atomic_fadd(rin1 + e1d[i], 1.0f);
    if (i < E2) {
      atomic_fadd(rout2 + e2s[i], 1.0f);
      atomic_fadd(rin2 + e2d[i], 1.0f);
    }
  }
}

// deg -> rsqrt(max(deg,1))
__global__ void k_degfinalize(float* __restrict__ p, int n) {
  int i = blockIdx.x * blockDim.x + threadIdx.x;
  int stride = gridDim.x * blockDim.x;
  for (; i < n; i += stride) p[i] = rsqrtf(fmaxf(p[i], 1.0f));
}

// ---------------- block1 edge aggregation ----------------
// agg1[dst] += lrelu(emb[nid1[src]]) * rout1[src]     (one wave32 per edge, 4 floats/lane)
__global__ void k_agg1(const float* __restrict__ emb, const int* __restrict__ nid1,
                       const int* __restrict__ es, const int* __restrict__ ed,
                       const float* __restrict__ rout1, float* __restrict__ agg1, int E) {
  int gwave = (blockIdx.x * blockDim.x + threadIdx.x) >> 5;
  int lane = threadIdx.x & 31;
  if (gwave >= E) return;
  int s = es[gwave];
  int d = ed[gwave];
  long node = nid1[s];
  float sc = rout1[s];
  float4 v = *(const float4*)(emb + node * DIM + lane * 4);
  float* dst = agg1 + (long)d * DIM + lane * 4;
  atomic_fadd(dst + 0, lrelu(v.x) * sc);
  atomic_fadd(dst + 1, lrelu(v.y) * sc);
  atomic_fadd(dst + 2, lrelu(v.z) * sc);
  atomic_fadd(dst + 3, lrelu(v.w) * sc);
}

// ---------------- block2 fused edge aggregation (h1 path + h2 path) ----------------
// agg2[dst] += lrelu(emb[nid2[src]]) * rout2[src] + hmid[src]   (hmid already carries rout2)
__global__ void k_agg2(const float* __restrict__ emb, const int* __restrict__ nid2,
                       const int* __restrict__ es, const int* __restrict__ ed,
                       const float* __restrict__ rout2, const float* __restrict__ hmid,
                       float* __restrict__ agg2, int E) {
  int gwave = (blockIdx.x * blockDim.x + threadIdx.x) >> 5;
  int lane = threadIdx.x & 31;
  if (gwave >= E) return;
  int s = es[gwave];
  int d = ed[gwave];
  long node = nid2[s];
  float sc = rout2[s];
  float4 v = *(const float4*)(emb + node * DIM + lane * 4);
  float4 hm = *(const float4*)(hmid + (long)s * DIM + lane * 4);
  float* dst = agg2 + (long)d * DIM + lane * 4;
  atomic_fadd(dst + 0, lrelu(v.x) * sc + hm.x);
  atomic_fadd(dst + 1, lrelu(v.y) * sc + hm.y);
  atomic_fadd(dst + 2, lrelu(v.z) * sc + hm.z);
  atomic_fadd(dst + 3, lrelu(v.w) * sc + hm.w);
}

// ---------------- fp32 WMMA GEMM: Out = epilogue(A[M,128] @ W[128,128]) ----------------
// mode 0: Out = lrelu(A@W * rowscale[r] + bias[c]) * postscale[r]          (GEMM0 + fuse)
// mode 1: Out = A@W * rowscale[r] + biasScale*bias[c] + emb[nid_dst[r]*D+c] (GEMM1 + h0)
// Block = 256 threads = 8 waves; each wave owns 16 rows, all 8 N-tiles (8 accumulators).
__global__ void k_gemm(const float* __restrict__ A, const float* __restrict__ Wm,
                       const float* __restrict__ bias, float biasScale,
                       const float* __restrict__ rowscale, const float* __restrict__ postscale,
                       const int* __restrict__ nid_dst, const float* __restrict__ emb,
                       float* __restrict__ Out, int M, int mode) {
  __shared__ float sW[DIM * DIM];              // 64 KB of the WGP's 320 KB LDS
  int tid = threadIdx.x;
  // cooperative stage of W (16384 floats = 4096 float4, 16 per thread)
#pragma unroll
  for (int i = 0; i < 16; ++i) {
    int idx = tid + i * 256;
    ((float4*)sW)[idx] = ((const float4*)Wm)[idx];
  }
  __syncthreads();

  int wave = tid >> 5;
  int lane = tid & 31;
  int m = lane & 15;        // row within A fragment
  int h = lane >> 4;        // lane half selects K pair
  int m0 = (blockIdx.x * 8 + wave) * 16;
  if (m0 >= M) return;      // wave-uniform exit (EXEC stays all-ones for WMMA)

  const float* arow = A + (long)(m0 + m) * DIM + 2 * h;

  v8f accs[8];
#pragma unroll
  for (int nt = 0; nt < 8; ++nt)
#pragma unroll
    for (int j = 0; j < 8; ++j) accs[nt][j] = 0.0f;

  // K loop: 32 steps of V_WMMA_F32_16X16X4_F32 per N-tile
  for (int kk = 0; kk < 32; ++kk) {
    int k0 = kk * 4;
    v2f a = *(const v2f*)(arow + k0);          // A[m][k0+2h .. k0+2h+1]
#pragma unroll
    for (int nt = 0; nt < 8; ++nt) {
      int col = nt * 16 + m;
      v2f b;
      b.x = sW[(k0 + 2 * h + 0) * DIM + col];  // B rows k0+2h, k0+2h+1
      b.y = sW[(k0 + 2 * h + 1) * DIM + col];
      accs[nt] = __builtin_amdgcn_wmma_f32_16x16x4_f32(
          false, a, false, b, (short)0, accs[nt], false, false);
    }
  }

  // epilogue: C/D layout -> row = m0 + v + 8h, col = nt*16 + m
  float rs_[8], rp_[8];
  long nidbase_[8];
#pragma unroll
  for (int v = 0; v < 8; ++v) {
    int r = m0 + v + 8 * h;
    rs_[v] = rowscale[r];
    if (mode == 0) rp_[v] = postscale[r];
    else nidbase_[v] = (long)nid_dst[r] * DIM;
  }
#pragma unroll
  for (int nt = 0; nt < 8; ++nt) {
    int col = nt * 16 + m;
    float bv = biasScale * bias[col];
#pragma unroll
    for (int v = 0; v < 8; ++v) {
      int r = m0 + v + 8 * h;
      float val = accs[nt][v] * rs_[v] + bv;
      if (mode == 0) val = lrelu(val) * rp_[v];
      else val += emb[nidbase_[v] + col];
      Out[(long)r * DIM + col] = val;
    }
  }
}

extern "C" void kernel_launch(void* const* d_in, const int* in_sizes, int n_in,
                              void* d_out, int out_size, void* d_ws, size_t ws_size,
                              hipStream_t stream) {
  const float* emb = (const float*)d_in[0];
  const float* W0  = (const float*)d_in[1];
  const float* b0  = (const float*)d_in[2];
  const float* W1  = (const float*)d_in[3];
  const float* b1  = (const float*)d_in[4];
  const int* nid_src1 = (const int*)d_in[5];
  const int* nid_src2 = (const int*)d_in[6];
  const int* nid_dst2 = (const int*)d_in[7];
  const int* e1_src = (const int*)d_in[8];
  const int* e1_dst = (const int*)d_in[9];
  const int* e2_src = (const int*)d_in[10];
  const int* e2_dst = (const int*)d_in[11];
  float* out = (float*)d_out;

  const int N1 = in_sizes[5];   // 495616
  const int N2 = in_sizes[6];   // 45056
  const int B  = in_sizes[7];   // 4096
  const int E1 = in_sizes[8];   // 450560
  const int E2 = in_sizes[10];  // 40960

  float* ws = (float*)d_ws;
  size_t o_rout1 = 0;
  size_t o_rin1  = o_rout1 + (size_t)N1;
  size_t o_rout2 = o_rin1  + (size_t)N2;
  size_t o_rin2  = o_rout2 + (size_t)N2;
  size_t o_agg1  = o_rin2  + (size_t)B;
  size_t o_agg2  = o_agg1  + (size_t)N2 * DIM;
  size_t o_hmid  = o_agg2  + (size_t)B * DIM;
  size_t zero_floats = o_hmid;       // degrees + agg1 + agg2 must start at 0

  // 1) zero scratch prefix
  k_zero<<<2048, 256, 0, stream>>>((float4*)ws, (long)(zero_floats / 4));

  // 2) degree counts, 3) rsqrt(max(deg,1))
  k_degcount<<<(E1 + 255) / 256, 256, 0, stream>>>(
      e1_src, e1_dst, e2_src, e2_dst, E1, E2,
      ws + o_rout1, ws + o_rin1, ws + o_rout2, ws + o_rin2);
  int ndeg = (int)o_agg1;
  k_degfinalize<<<(ndeg + 255) / 256, 256, 0, stream>>>(ws, ndeg);

  // 4) block1 edge gather/scatter: agg1 = sum_e lrelu(emb[nid1[s]])*rout1[s]
  k_agg1<<<(E1 + 7) / 8, 256, 0, stream>>>(
      emb, nid_src1, e1_src, e1_dst, ws + o_rout1, ws + o_agg1, E1);

  // 5) GEMM0 (WMMA fp32): hmid = lrelu(agg1@W0 * rin1 + b0) * rout2
  k_gemm<<<(N2 + 127) / 128, 256, 0, stream>>>(
      ws + o_agg1, W0, b0, 1.0f, ws + o_rin1, ws + o_rout2,
      nullptr, nullptr, ws + o_hmid, N2, 0);

  // 6) block2 fused aggregation of h1+h2 paths into agg2
  k_agg2<<<(E2 + 7) / 8, 256, 0, stream>>>(
      emb, nid_src2, e2_src, e2_dst, ws + o_rout2, ws + o_hmid, ws + o_agg2, E2);

  // 7) GEMM1 (WMMA fp32): out = agg2@W1 * rin2 + 2*b1 + emb[nid_dst2]  (== h0+h1+h2)
  k_gemm<<<(B + 127) / 128, 256, 0, stream>>>(
      ws + o_agg2, W1, b1, 2.0f, ws + o_rin2, nullptr,
      nid_dst2, emb, out, B, 1);
}